// Model_2044404433417
// MI455X (gfx1250) — compile-verified
//
#include <hip/hip_runtime.h>
#include <math.h>
#include <stdint.h>

typedef __attribute__((ext_vector_type(2))) float v2f;
typedef __attribute__((ext_vector_type(8))) float v8f;
typedef __attribute__((ext_vector_type(4))) unsigned int u32x4;
typedef __attribute__((ext_vector_type(8))) int i32x8;
typedef __attribute__((ext_vector_type(4))) int i32x4;

#define DMODEL 128
#define NHEADS 8
#define DH     16
#define BB     8
#define PP     512
#define SS     1024
#define SCALE_ 0.25f      // DH^-0.5

#define KROW   18         // LDS row stride (floats) for K/V slice: 16 + TDM pad of 2
#define SROW   1028       // LDS row stride (floats) for score stripe: 1024 + 4 pad

#if defined(__HIP_DEVICE_COMPILE__) && __has_builtin(__builtin_amdgcn_tensor_load_to_lds)
#define USE_TDM 1
#else
#define USE_TDM 0
#endif

// ---------------------------------------------------------------------------
// Generic projection GEMM: C[M,128] = A[M,128] @ W[128,128]^T + bias[128]
// One wave per 16x16 output tile, V_WMMA_F32_16X16X4_F32 over K=128.
// ---------------------------------------------------------------------------
__global__ void proj_gemm(const float* __restrict__ A, const float* __restrict__ W,
                          const float* __restrict__ bias, float* __restrict__ C) {
    const int lane = threadIdx.x & 31;
    const int wave = threadIdx.x >> 5;
    const int tile = blockIdx.x * (blockDim.x >> 5) + wave;
    const int tm = tile >> 3;
    const int tn = tile & 7;
    const int l15 = lane & 15;
    const int kg  = (lane >> 4) << 1;

    const float* arow = A + (size_t)(tm * 16 + l15) * DMODEL;
    const float* wrow = W + (size_t)(tn * 16 + l15) * DMODEL;   // B[k][n] = W[n][k]

    v8f acc = {};
#pragma unroll 8
    for (int k = 0; k < DMODEL; k += 4) {
        v2f a, b;
        a.x = arow[k + kg];     a.y = arow[k + kg + 1];
        b.x = wrow[k + kg];     b.y = wrow[k + kg + 1];
        acc = __builtin_amdgcn_wmma_f32_16x16x4_f32(false, a, false, b,
                                                    (short)0, acc, false, false);
    }

    const int row0 = tm * 16 + ((lane >> 4) << 3);
    const int col  = tn * 16 + l15;
    const float bc = bias[col];
#pragma unroll
    for (int r = 0; r < 8; ++r)
        C[(size_t)(row0 + r) * DMODEL + col] = acc[r] + bc;
}

// ---------------------------------------------------------------------------
// TDM: stage one head-slice (SS rows x 16 floats, row stride 128 floats) into
// LDS, compacted to KROW-float rows via the TDM pad feature
// (pad_interval = 16 dwords -> code 3, pad_amount = 2 dwords -> code 1).
// ---------------------------------------------------------------------------
#if USE_TDM
__device__ __forceinline__ void tdm_stage_slice(const float* gptr, float* lds_dst) {
    const unsigned long long ga = (unsigned long long)(uintptr_t)gptr;
    const unsigned ldsa = (unsigned)(uintptr_t)lds_dst;   // addr[31:0] = LDS offset
    u32x4 g0;
    g0[0] = 1u;                                           // count = 1 (valid D#)
    g0[1] = ldsa;                                         // lds_addr
    g0[2] = (unsigned)(ga & 0xFFFFFFFFu);                 // global_addr[31:0]
    g0[3] = (unsigned)((ga >> 32) & 0x01FFFFFFu) | 0x80000000u;  // addr[56:32] | type=2
    i32x8 g1;
    g1[0] = (int)((2u << 16) | (1u << 20) | (3u << 22) | (1u << 25));
    //            data=4B      pad_en       ivl=16dw      amt=2dw
    g1[1] = (int)(16u << 16);      // tensor_dim0 = 16 (bits 79:48, lo16)
    g1[2] = (int)(1024u << 16);    // tensor_dim1 = 1024 (lo16); dim0 hi16 = 0
    g1[3] = (int)(16u << 16);      // tile_dim0 = 16; dim1 hi16 = 0
    g1[4] = 1024;                  // tile_dim1 = 1024, tile_dim2 = 0
    g1[5] = DMODEL;                // tensor_dim0_stride = 128 elements (lo32)
    g1[6] = 0;
    g1[7] = 0;
    const i32x4 z4 = {0, 0, 0, 0};
    const i32x8 z8 = {0, 0, 0, 0, 0, 0, 0, 0};
    __builtin_amdgcn_tensor_load_to_lds(g0, g1, z4, z4, z8, 0);
}
#endif

__device__ __forceinline__ void stage_slice_fallback(const float* __restrict__ g,
                                                     float* __restrict__ dst, int tid) {
    for (int i = tid; i < SS * DH; i += 256) {
        const int s = i >> 4, d = i & 15;
        dst[s * KROW + d] = g[(size_t)s * DMODEL + d];
    }
}

// ---------------------------------------------------------------------------
// Fused scores + residual + gate + masked softmax + (attn @ v) per
// (b, h, 16-row P-tile). 8 waves / 256 threads per workgroup.
//   phase 1: scores (WMMA, K from LDS) -> raw_scores global + LDS stripe
//   phase 2: masked softmax from LDS -> attn global + LDS (V TDM in flight)
//   phase 3: attn @ v (WMMA, both operands from LDS), K split across waves
// ---------------------------------------------------------------------------
__global__ void fused_attn(const float* __restrict__ qbuf, const float* __restrict__ kbuf,
                           const float* __restrict__ vbuf, const float* __restrict__ prev,
                           const float* __restrict__ gate,
                           float* __restrict__ raw, float* __restrict__ attn,
                           float* __restrict__ outh) {
    extern __shared__ float smem[];
    float* ldsS = smem;                       // 16 x SROW    (score stripe)
    float* ldsK = ldsS + 16 * SROW;           // SS x KROW    (K, then V, slice)
    float* part = ldsK + SS * KROW;           // 8 x 256      (AV partial tiles)

    const int tid  = threadIdx.x;
    const int lane = tid & 31;
    const int wave = tid >> 5;
    const int wg   = blockIdx.x;              // B*H*32 workgroups
    const int tm   = wg & 31;
    const int bh   = wg >> 5;
    const int h    = bh & 7;
    const int b    = bh >> 3;
    const int p0   = tm * 16;
    const int l15  = lane & 15;
    const int kg   = (lane >> 4) << 1;
    const int pc   = p0 >> 6;                 // pred channel (uniform in WG)

    // ---- stage K head-slice into LDS -------------------------------------
    const float* gk = kbuf + (size_t)(b * SS) * DMODEL + h * DH;
#if USE_TDM
    if (wave == 0) {
        tdm_stage_slice(gk, ldsK);
        __builtin_amdgcn_s_wait_tensorcnt(0);
    }
#else
    stage_slice_fallback(gk, ldsK, tid);
#endif
    __syncthreads();

    // ---- per-wave A operand (q rows p0..p0+15, reused for all 8 s-tiles) --
    const float* qrow = qbuf + (size_t)(b * PP + p0 + l15) * DMODEL + h * DH;
    v2f aop[4];
#pragma unroll
    for (int j = 0; j < 4; ++j) {
        aop[j].x = qrow[4 * j + kg];
        aop[j].y = qrow[4 * j + kg + 1];
    }

    // ---- phase 1: scores --------------------------------------------------
    const int prow0 = (lane >> 4) << 3;
#pragma unroll
    for (int t = 0; t < 8; ++t) {
        const int tn = wave * 8 + t;
        const int s0 = tn * 16;
        v8f acc = {};
#pragma unroll
        for (int k = 0; k < DH; k += 4) {
            v2f bm;
            const int kro = (s0 + l15) * KROW + k + kg;
            bm.x = ldsK[kro];
            bm.y = ldsK[kro + 1];
            acc = __builtin_amdgcn_wmma_f32_16x16x4_f32(false, aop[k >> 2], false, bm,
                                                        (short)0, acc, false, false);
        }
        const int sc = s0 >> 7;                                   // seq channel
        const float g = (gate[pc * 8 + sc] >= 0.0f) ? 1.0f : 0.0f;  // STE forward
#pragma unroll
        for (int r = 0; r < 8; ++r) {
            const int pl = prow0 + r;
            const size_t idx = (((size_t)(b * PP + p0 + pl)) * NHEADS + h) * SS + s0 + l15;
            float v = acc[r] * SCALE_ + prev[idx];
            v = __builtin_isfinite(v) ? v * g : v;
            raw[idx] = v;
            ldsS[pl * SROW + s0 + l15] = v;
        }
    }
    __syncthreads();

    // ---- kick off async V staging (overlaps with softmax) -----------------
    const float* gv = vbuf + (size_t)(b * SS) * DMODEL + h * DH;
#if USE_TDM
    if (wave == 0) tdm_stage_slice(gv, ldsK);
#else
    stage_slice_fallback(gv, ldsK, tid);
#endif

    // ---- phase 2: masked softmax, 2 rows per wave -------------------------
#pragma unroll
    for (int rr = 0; rr < 2; ++rr) {
        const int row = wave * 2 + rr;
        const float* srow = ldsS + row * SROW;
        float ev[SS / 32];
        float mmax = -__builtin_inff();
#pragma unroll 4
        for (int i = 0; i < SS / 32; ++i) {
            const int s = lane + 32 * i;
            const float x = srow[s];
            ev[i] = x;
            if ((s >> 7) != pc) mmax = fmaxf(mmax, x);
        }
#pragma unroll
        for (int o = 16; o > 0; o >>= 1)
            mmax = fmaxf(mmax, __shfl_xor(mmax, o, 32));
        float z = 0.0f;
#pragma unroll 4
        for (int i = 0; i < SS / 32; ++i) {
            const int s = lane + 32 * i;
            const float e = ((s >> 7) != pc) ? expf(ev[i] - mmax) : 0.0f;
            ev[i] = e;
            z += e;
        }
#pragma unroll
        for (int o = 16; o > 0; o >>= 1)
            z += __shfl_xor(z, o, 32);
        const float inv = 1.0f / fmaxf(z, 1e-9f);
        float* wout = attn + (((size_t)(b * PP + p0 + row)) * NHEADS + h) * SS;
        float* sw   = ldsS + row * SROW;
#pragma unroll 4
        for (int i = 0; i < SS / 32; ++i) {
            const int s = lane + 32 * i;
            const float val = ev[i] * inv;
            wout[s] = val;
            sw[s]   = val;
        }
    }
#if USE_TDM
    if (wave == 0) __builtin_amdgcn_s_wait_tensorcnt(0);
#endif
    __syncthreads();

    // ---- phase 3: attn @ v, K-range [wave*128, wave*128+128) --------------
    {
        const int kbase = wave * 128;
        v8f acc = {};
#pragma unroll 8
        for (int kk = 0; kk < 128; kk += 4) {
            const int kidx = kbase + kk;
            v2f a, bm;
            const int ao = l15 * SROW + kidx + kg;
            a.x  = ldsS[ao];
            a.y  = ldsS[ao + 1];
            bm.x = ldsK[(kidx + kg) * KROW + l15];
            bm.y = ldsK[(kidx + kg + 1) * KROW + l15];
            acc = __builtin_amdgcn_wmma_f32_16x16x4_f32(false, a, false, bm,
                                                        (short)0, acc, false, false);
        }
#pragma unroll
        for (int r = 0; r < 8; ++r)
            part[wave * 256 + (prow0 + r) * 16 + l15] = acc[r];
    }
    __syncthreads();

    // ---- combine partials, write head-concatenated output -----------------
    float ssum = 0.0f;
#pragma unroll
    for (int w = 0; w < 8; ++w) ssum += part[w * 256 + tid];
    const int rowl = tid >> 4, col = tid & 15;
    outh[(size_t)(b * PP + p0 + rowl) * DMODEL + h * DH + col] = ssum;
}

// ---------------------------------------------------------------------------
extern "C" void kernel_launch(void* const* d_in, const int* in_sizes, int n_in,
                              void* d_out, int out_size, void* d_ws, size_t ws_size,
                              hipStream_t stream) {
    const float* Q    = (const float*)d_in[0];
    const float* K    = (const float*)d_in[1];
    const float* V    = (const float*)d_in[2];
    const float* prev = (const float*)d_in[3];
    // d_in[4] = attn_mask: derivable from indices, not read
    const float* Wq = (const float*)d_in[5];  const float* bq = (const float*)d_in[6];
    const float* Wk = (const float*)d_in[7];  const float* bk = (const float*)d_in[8];
    const float* Wv = (const float*)d_in[9];  const float* bv = (const float*)d_in[10];
    const float* Wo = (const float*)d_in[11]; const float* bo = (const float*)d_in[12];
    const float* gate = (const float*)d_in[13];

    float* out  = (float*)d_out;                                   // (B,P,128)
    float* attn = out  + (size_t)BB * PP * DMODEL;                 // (B,P,H,S)
    float* raw  = attn + (size_t)BB * PP * NHEADS * SS;            // (B,P,H,S)

    float* qb = (float*)d_ws;                                      // (B*P,128)
    float* kb = qb + (size_t)BB * PP * DMODEL;                     // (B*S,128)
    float* vb = kb + (size_t)BB * SS * DMODEL;                     // (B*S,128)
    float* oh = vb + (size_t)BB * SS * DMODEL;                     // (B*P,128)

    const dim3 blk(256);

    proj_gemm<<<256, blk, 0, stream>>>(Q, Wq, bq, qb);   // M=4096 -> 2048 tiles
    proj_gemm<<<512, blk, 0, stream>>>(K, Wk, bk, kb);   // M=8192 -> 4096 tiles
    proj_gemm<<<512, blk, 0, stream>>>(V, Wv, bv, vb);   // M=8192 -> 4096 tiles

    const size_t lds_bytes = (size_t)(16 * SROW + SS * KROW + 8 * 256) * sizeof(float);
    fused_attn<<<BB * NHEADS * (PP / 16), blk, lds_bytes, stream>>>(
        qb, kb, vb, prev, gate, raw, attn, oh);          // 2048 workgroups

    proj_gemm<<<256, blk, 0, stream>>>(oh, Wo, bo, out); // M=4096 -> 2048 tiles
}